// classifer_23725399343149
// MI455X (gfx1250) — compile-verified
//
#include <hip/hip_runtime.h>
#include <hip/hip_bf16.h>

// Problem constants (from the reference)
#define BATCH 256
#define CH    512
#define HW    784          // 28*28
#define HW4   196          // HW/4 (float4 count per row)
#define NCLS  20

typedef __attribute__((ext_vector_type(2))) float v2f;
typedef __attribute__((ext_vector_type(4))) float v4f;
typedef __attribute__((ext_vector_type(8))) float v8f;

// ---------------------------------------------------------------------------
// Kernel 1: AvgPool2d(28) -> pooled[B*C]. One wave32 per (b,c) row of 784 f32.
// Streams 411 MB with non-temporal float4 loads (buffer >> 192MB L2).
// ---------------------------------------------------------------------------
__global__ __launch_bounds__(256) void pool_kernel(const v4f* __restrict__ x4,
                                                   float* __restrict__ pooled) {
    const int lane = threadIdx.x & 31;
    const int wave = threadIdx.x >> 5;
    const int row  = blockIdx.x * 8 + wave;          // row in [0, B*C)
    const v4f* p = x4 + (size_t)row * HW4;

    float sum = 0.0f;
    #pragma unroll
    for (int i = lane; i < HW4; i += 32) {
        v4f v = __builtin_nontemporal_load(&p[i]);
        sum += (v.x + v.y) + (v.z + v.w);
    }
    // wave32 tree reduction
    #pragma unroll
    for (int off = 16; off > 0; off >>= 1)
        sum += __shfl_down(sum, off, 32);
    if (lane == 0)
        pooled[row] = sum * (1.0f / (float)HW);
}

// ---------------------------------------------------------------------------
// Kernel 2: logits[256,20] = pooled[256,512] @ W^T[512,20] + b
// fp32 WMMA: V_WMMA_F32_16X16X4_F32, one wave per 16x16 output tile.
// Grid: (M/16, ceil(N/16)) = (16, 2); block = 32 (one wave, EXEC all ones).
// A frag (16x4 f32): lane<16 -> K={k0,k0+1}; lane>=16 -> K={k0+2,k0+3}, M=lane&15.
// B frag (4x16 f32): same K split, N=lane&15.
// C/D (16x16 f32): VGPR r -> M = r + (lane<16 ? 0 : 8), N = lane&15.
// N=20..31 lanes load a clamped (valid) W row and scale by 0 — keeps the
// K-loop branch-free (no per-iteration exec save/restore).
// ---------------------------------------------------------------------------
__global__ __launch_bounds__(32) void logits_wmma_kernel(const float* __restrict__ pooled,
                                                         const float* __restrict__ W,
                                                         const float* __restrict__ bias,
                                                         float* __restrict__ logits) {
    const int lane  = threadIdx.x;          // 0..31
    const int mtile = blockIdx.x;           // 0..15
    const int ntile = blockIdx.y;           // 0..1
    const int mrow  = mtile * 16 + (lane & 15);
    const int ncol  = ntile * 16 + (lane & 15);
    const int koff  = (lane < 16) ? 0 : 2;
    const int ncolc = (ncol < NCLS) ? ncol : (NCLS - 1);     // clamped, always valid
    const float wmask = (ncol < NCLS) ? 1.0f : 0.0f;

    const float* __restrict__ Ap = pooled + (size_t)mrow * CH + koff;
    const float* __restrict__ Wp = W + (size_t)ncolc * CH + koff;   // B[k][n] = W[n][k]

    v8f acc = {};
    for (int k0 = 0; k0 < CH; k0 += 4) {
        v2f a, bm;
        a.x  = Ap[k0];
        a.y  = Ap[k0 + 1];
        bm.x = Wp[k0] * wmask;
        bm.y = Wp[k0 + 1] * wmask;
        acc = __builtin_amdgcn_wmma_f32_16x16x4_f32(
            /*neg_a=*/false, a, /*neg_b=*/false, bm,
            /*c_mod=*/(short)0, acc, /*reuse_a=*/false, /*reuse_b=*/false);
    }

    if (ncol < NCLS) {
        const int mbase = mtile * 16 + ((lane < 16) ? 0 : 8);
        const float bv = bias[ncol];
        #pragma unroll
        for (int r = 0; r < 8; ++r)
            logits[(mbase + r) * NCLS + ncol] = acc[r] + bv;
    }
}

// ---------------------------------------------------------------------------
// Kernel 3: reweight[256,512] = float(label[256,20]) @ W[20,512]
// Same fp32 WMMA path; K = 20 -> 5 k-steps of 4. Grid (16, 32), block 32.
// ---------------------------------------------------------------------------
__global__ __launch_bounds__(32) void reweight_wmma_kernel(const int* __restrict__ label,
                                                           const float* __restrict__ W,
                                                           float* __restrict__ rw) {
    const int lane  = threadIdx.x;          // 0..31
    const int mtile = blockIdx.x;           // 0..15  (M = 256)
    const int ntile = blockIdx.y;           // 0..31  (N = 512)
    const int mrow  = mtile * 16 + (lane & 15);
    const int ncol  = ntile * 16 + (lane & 15);
    const int koff  = (lane < 16) ? 0 : 2;

    const int*   __restrict__ Lp = label + (size_t)mrow * NCLS + koff;
    const float* __restrict__ Wp = W + (size_t)koff * CH + ncol;     // B[k][n] = W[k][n]

    v8f acc = {};
    #pragma unroll
    for (int k0 = 0; k0 < NCLS; k0 += 4) {
        v2f a, bm;
        a.x  = (float)Lp[k0];
        a.y  = (float)Lp[k0 + 1];
        bm.x = Wp[(size_t)k0 * CH];
        bm.y = Wp[(size_t)(k0 + 1) * CH];
        acc = __builtin_amdgcn_wmma_f32_16x16x4_f32(
            false, a, false, bm, (short)0, acc, false, false);
    }

    const int mbase = mtile * 16 + ((lane < 16) ? 0 : 8);
    #pragma unroll
    for (int r = 0; r < 8; ++r)
        rw[(size_t)(mbase + r) * CH + ncol] = acc[r];
}

// ---------------------------------------------------------------------------
// Kernel 4: out_map = fmap * reweight[b,c]  (broadcast over 28x28)
// Dominant stream: 411 MB NT read + 411 MB NT write. reweight (512 KB) stays
// cache-resident. Exact mapping: B*C*196 float4 elements, 256 threads/block.
// ---------------------------------------------------------------------------
__global__ __launch_bounds__(256) void scale_kernel(const v4f* __restrict__ fmap4,
                                                    const float* __restrict__ rw,
                                                    v4f* __restrict__ out4) {
    const unsigned idx = blockIdx.x * 256u + threadIdx.x;   // < B*C*196 exactly
    const unsigned row = idx / (unsigned)HW4;               // (b*C + c)
    const float s = rw[row];
    v4f v = __builtin_nontemporal_load(&fmap4[idx]);
    v *= s;
    __builtin_nontemporal_store(v, &out4[idx]);
}

// ---------------------------------------------------------------------------
extern "C" void kernel_launch(void* const* d_in, const int* in_sizes, int n_in,
                              void* d_out, int out_size, void* d_ws, size_t ws_size,
                              hipStream_t stream) {
    const float* x     = (const float*)d_in[0];   // [B,C,28,28]
    const float* fmap  = (const float*)d_in[1];   // [B,C,28,28]
    const int*   label = (const int*)d_in[2];     // [B,NCLS]
    const float* W     = (const float*)d_in[3];   // [NCLS,C]
    const float* bias  = (const float*)d_in[4];   // [NCLS]

    float* logits  = (float*)d_out;                          // [B,NCLS] = 5120 f32
    float* out_map = (float*)d_out + (size_t)BATCH * NCLS;   // [B,C,28,28]

    // Workspace: pooled (512 KB) then reweight (512 KB)
    float* pooled = (float*)d_ws;
    float* rw     = pooled + (size_t)BATCH * CH;

    // 1) pooled = mean(x, HW)   -- one wave per row, 8 waves per block
    pool_kernel<<<(BATCH * CH) / 8, 256, 0, stream>>>((const v4f*)x, pooled);

    // 2) logits = pooled @ W^T + b   (fp32 WMMA)
    logits_wmma_kernel<<<dim3(BATCH / 16, 2), 32, 0, stream>>>(pooled, W, bias, logits);

    // 3) reweight = label @ W        (fp32 WMMA)
    reweight_wmma_kernel<<<dim3(BATCH / 16, CH / 16), 32, 0, stream>>>(label, W, rw);

    // 4) out_map = fmap * reweight broadcast
    const unsigned total4 = (unsigned)BATCH * CH * HW4;      // 25,690,112
    scale_kernel<<<total4 / 256, 256, 0, stream>>>((const v4f*)fmap, rw, (v4f*)out_map);
}